// GNNMarketModel_59854664237267
// MI455X (gfx1250) — compile-verified
//
#include <hip/hip_runtime.h>
#include <hip/hip_bf16.h>

#define N_NODES 100000
#define N_EDGES 3200000
#define IN_DIM  128
#define HID     64
#define N_GRAPHS 64
#define BN_EPS  1e-5f

typedef float v2f __attribute__((ext_vector_type(2)));
typedef float v8f __attribute__((ext_vector_type(8)));

// ---------------------------------------------------------------------------
// init: deg = 1.0 (self-loop weight), zero the small accumulators
// ---------------------------------------------------------------------------
__global__ void k_init(float* __restrict__ deg, float* __restrict__ small, int nsmall) {
    int i = blockIdx.x * blockDim.x + threadIdx.x;
    if (i < N_NODES) deg[i] = 1.0f;
    if (i < nsmall)  small[i] = 0.0f;
}

// deg[col] += w  (target-node in-degree)
__global__ void k_deg(const int* __restrict__ ei, const float* __restrict__ w,
                      float* __restrict__ deg) {
    int e = blockIdx.x * blockDim.x + threadIdx.x;
    if (e < N_EDGES) atomicAdd(&deg[ei[N_EDGES + e]], w[e]);
}

__global__ void k_dinv(const float* __restrict__ deg, float* __restrict__ dinv) {
    int i = blockIdx.x * blockDim.x + threadIdx.x;
    if (i < N_NODES) {
        float d = deg[i];
        dinv[i] = (d > 0.0f) ? rsqrtf(d) : 0.0f;
    }
}

__global__ void k_norm(const int* __restrict__ ei, const float* __restrict__ w,
                       const float* __restrict__ dinv, float* __restrict__ norm) {
    int e = blockIdx.x * blockDim.x + threadIdx.x;
    if (e < N_EDGES) {
        int r = ei[e];
        int c = ei[N_EDGES + e];
        norm[e] = dinv[r] * w[e] * dinv[c];
    }
}

// ---------------------------------------------------------------------------
// GEMM via V_WMMA_F32_16X16X4_F32: C[N x 64] = A[N x K] * B[K x 64]
// One wave computes a 16-row x 64-col tile (4 accumulators).
// A frag (16x4 f32): lanes 0-15 hold M=lane, K = k+0 / k+1;
//                    lanes 16-31 hold M=lane-16, K = k+2 / k+3.
// B frag (4x16 f32): VGPR j, lanes 0-15 = row K=k+khalf+j, N=lane;
//                    lanes 16-31 = row K=k+2+j (khalf folds this in).
// C/D (16x16 f32):   VGPR r, lanes 0-15 -> M=r, lanes 16-31 -> M=r+8, N=lane&15.
// ---------------------------------------------------------------------------
__global__ __launch_bounds__(128)
void k_gemm_wmma(const float* __restrict__ A, const float* __restrict__ B,
                 float* __restrict__ C, int nrows, int K) {
    const int wave = threadIdx.x >> 5;
    const int lane = threadIdx.x & 31;
    const int tile = blockIdx.x * 4 + wave;
    const int row0 = tile * 16;
    if (row0 >= nrows) return;

    const int m     = lane & 15;        // row within A tile / col within B tile
    const int khalf = (lane >> 4) * 2;  // 0 for lanes 0-15, 2 for lanes 16-31

    v8f acc0 = {}, acc1 = {}, acc2 = {}, acc3 = {};

    const float* arow = A + (size_t)(row0 + m) * K;
    for (int k = 0; k < K; k += 4) {
        v2f a;
        a[0] = arow[k + khalf + 0];
        a[1] = arow[k + khalf + 1];

        const float* b0 = B + (size_t)(k + khalf + 0) * HID;
        const float* b1 = B + (size_t)(k + khalf + 1) * HID;
        v2f bb0, bb1, bb2, bb3;
        bb0[0] = b0[m +  0]; bb0[1] = b1[m +  0];
        bb1[0] = b0[m + 16]; bb1[1] = b1[m + 16];
        bb2[0] = b0[m + 32]; bb2[1] = b1[m + 32];
        bb3[0] = b0[m + 48]; bb3[1] = b1[m + 48];

        acc0 = __builtin_amdgcn_wmma_f32_16x16x4_f32(false, a, false, bb0, (short)0, acc0, false, false);
        acc1 = __builtin_amdgcn_wmma_f32_16x16x4_f32(false, a, false, bb1, (short)0, acc1, false, false);
        acc2 = __builtin_amdgcn_wmma_f32_16x16x4_f32(false, a, false, bb2, (short)0, acc2, false, false);
        acc3 = __builtin_amdgcn_wmma_f32_16x16x4_f32(false, a, false, bb3, (short)0, acc3, false, false);
    }

    const int mrow = row0 + (lane >> 4) * 8;
    const int n    = lane & 15;
    #pragma unroll
    for (int r = 0; r < 8; ++r) {
        float* crow = C + (size_t)(mrow + r) * HID + n;
        crow[ 0] = acc0[r];
        crow[16] = acc1[r];
        crow[32] = acc2[r];
        crow[48] = acc3[r];
    }
}

// agg[i][f] = h[i][f] * dinv[i]^2   (self-loop contribution, full overwrite)
__global__ void k_selfinit(const float* __restrict__ h, const float* __restrict__ dinv,
                           float* __restrict__ agg) {
    size_t idx = (size_t)blockIdx.x * blockDim.x + threadIdx.x;
    if (idx < (size_t)N_NODES * HID) {
        int i = (int)(idx >> 6);
        float d = dinv[i];
        agg[idx] = h[idx] * d * d;
    }
}

// scatter-add: 16 lanes per edge, float4 per lane
__global__ void k_scatter(const float* __restrict__ h, const int* __restrict__ ei,
                          const float* __restrict__ norm, float* __restrict__ agg) {
    size_t t  = (size_t)blockIdx.x * blockDim.x + threadIdx.x;
    int    e  = (int)(t >> 4);
    int    f4 = ((int)t & 15) * 4;
    if (e >= N_EDGES) return;
    int   r  = ei[e];
    int   c  = ei[N_EDGES + e];
    float nw = norm[e];
    const float4 hv = *(const float4*)(h + (size_t)r * HID + f4);
    float* dst = agg + (size_t)c * HID + f4;
    atomicAdd(dst + 0, hv.x * nw);
    atomicAdd(dst + 1, hv.y * nw);
    atomicAdd(dst + 2, hv.z * nw);
    atomicAdd(dst + 3, hv.w * nw);
}

// BN stats pass: per-feature sum and sum-of-squares of (agg + bias)
__global__ __launch_bounds__(256)
void k_bnstats(const float* __restrict__ agg, const float* __restrict__ b,
               float* __restrict__ s, float* __restrict__ q) {
    __shared__ float ls[4][HID];
    __shared__ float lq[4][HID];
    const int f   = threadIdx.x & 63;
    const int sub = threadIdx.x >> 6;
    const int base = blockIdx.x * 256;
    const float bf = b[f];
    float sum = 0.0f, sq = 0.0f;
    for (int i = sub; i < 256; i += 4) {
        int node = base + i;
        if (node < N_NODES) {
            float v = agg[(size_t)node * HID + f] + bf;
            sum += v;
            sq  += v * v;
        }
    }
    ls[sub][f] = sum;
    lq[sub][f] = sq;
    __syncthreads();
    if (sub == 0) {
        sum = ls[0][f] + ls[1][f] + ls[2][f] + ls[3][f];
        sq  = lq[0][f] + lq[1][f] + lq[2][f] + lq[3][f];
        atomicAdd(&s[f], sum);
        atomicAdd(&q[f], sq);
    }
}

// BN apply + ReLU: z = relu(gamma*(v-mean)*rsqrt(var+eps)+beta)
__global__ void k_bnapply(const float* __restrict__ agg, const float* __restrict__ b,
                          const float* __restrict__ s, const float* __restrict__ q,
                          const float* __restrict__ g, const float* __restrict__ be,
                          float* __restrict__ z) {
    size_t idx = (size_t)blockIdx.x * blockDim.x + threadIdx.x;
    if (idx < (size_t)N_NODES * HID) {
        int f = (int)idx & 63;
        const float invn = 1.0f / (float)N_NODES;
        float mean = s[f] * invn;
        float var  = q[f] * invn - mean * mean;
        float v    = agg[idx] + b[f];
        float y    = g[f] * (v - mean) * rsqrtf(var + BN_EPS) + be[f];
        z[idx] = fmaxf(y, 0.0f);
    }
}

// global mean pool accumulation
__global__ void k_pool(const float* __restrict__ z, const int* __restrict__ batch,
                       float* __restrict__ pooled, float* __restrict__ counts) {
    size_t idx = (size_t)blockIdx.x * blockDim.x + threadIdx.x;
    if (idx < (size_t)N_NODES * HID) {
        int i = (int)(idx >> 6);
        int f = (int)idx & 63;
        int gidx = batch[i];
        atomicAdd(&pooled[(size_t)gidx * HID + f], z[idx]);
        if (f == 0) atomicAdd(&counts[gidx], 1.0f);
    }
}

// out[g] = (pooled[g]/max(count,1)) . Wout + bout
__global__ void k_out(const float* __restrict__ pooled, const float* __restrict__ counts,
                      const float* __restrict__ Wout, const float* __restrict__ bout,
                      float* __restrict__ out) {
    int gidx = threadIdx.x;
    if (gidx < N_GRAPHS) {
        float c   = fmaxf(counts[gidx], 1.0f);
        float inv = 1.0f / c;
        float acc = 0.0f;
        #pragma unroll
        for (int f = 0; f < HID; ++f)
            acc += pooled[(size_t)gidx * HID + f] * inv * Wout[f];
        out[gidx] = acc + bout[0];
    }
}

extern "C" void kernel_launch(void* const* d_in, const int* in_sizes, int n_in,
                              void* d_out, int out_size, void* d_ws, size_t ws_size,
                              hipStream_t stream) {
    const float* x    = (const float*)d_in[0];
    const int*   ei   = (const int*)  d_in[1];
    const float* ew   = (const float*)d_in[2];
    const int*   bat  = (const int*)  d_in[3];
    const float* W1   = (const float*)d_in[4];
    const float* b1   = (const float*)d_in[5];
    const float* W2   = (const float*)d_in[6];
    const float* b2   = (const float*)d_in[7];
    const float* g1   = (const float*)d_in[8];
    const float* be1  = (const float*)d_in[9];
    const float* g2   = (const float*)d_in[10];
    const float* be2  = (const float*)d_in[11];
    const float* Wout = (const float*)d_in[12];
    const float* bout = (const float*)d_in[13];
    float* out = (float*)d_out;

    // workspace layout (floats)
    float* ws     = (float*)d_ws;
    float* deg    = ws;
    float* dinv   = deg  + N_NODES;
    float* norm   = dinv + N_NODES;
    float* bufA   = norm + N_EDGES;                 // h1 -> z1 -> agg2
    float* bufB   = bufA + (size_t)N_NODES * HID;   // agg1 -> h2 -> z2
    float* s1     = bufB + (size_t)N_NODES * HID;
    float* q1     = s1 + HID;
    float* s2     = q1 + HID;
    float* q2     = s2 + HID;
    float* pooled = q2 + HID;                       // 64*64
    float* counts = pooled + (size_t)N_GRAPHS * HID;
    const int nsmall = 4 * HID + N_GRAPHS * HID + N_GRAPHS;

    const int TB = 256;
    const int blkN  = (N_NODES + TB - 1) / TB;                 // 391
    const int blkE  = (N_EDGES + TB - 1) / TB;                 // 12500
    const int blkNH = (int)(((size_t)N_NODES * HID + TB - 1) / TB); // 25000
    const int blkSC = (int)(((size_t)N_EDGES * 16 + TB - 1) / TB);  // 200000
    const int blkG  = (N_NODES / 16 + 3) / 4;                  // 1563 (6250 tiles)

    // --- normalization coefficients ---
    k_init  <<<blkN,  TB, 0, stream>>>(deg, s1, nsmall);
    k_deg   <<<blkE,  TB, 0, stream>>>(ei, ew, deg);
    k_dinv  <<<blkN,  TB, 0, stream>>>(deg, dinv);
    k_norm  <<<blkE,  TB, 0, stream>>>(ei, ew, dinv, norm);

    // --- layer 1: GCNConv -> BN -> ReLU ---
    k_gemm_wmma<<<blkG, 128, 0, stream>>>(x, W1, bufA, N_NODES, IN_DIM); // h1 = x@W1
    k_selfinit <<<blkNH, TB, 0, stream>>>(bufA, dinv, bufB);             // agg1 = h1*dinv^2
    k_scatter  <<<blkSC, TB, 0, stream>>>(bufA, ei, norm, bufB);         // agg1 += msgs
    k_bnstats  <<<blkN,  TB, 0, stream>>>(bufB, b1, s1, q1);
    k_bnapply  <<<blkNH, TB, 0, stream>>>(bufB, b1, s1, q1, g1, be1, bufA); // z1

    // --- layer 2 ---
    k_gemm_wmma<<<blkG, 128, 0, stream>>>(bufA, W2, bufB, N_NODES, HID); // h2 = z1@W2
    k_selfinit <<<blkNH, TB, 0, stream>>>(bufB, dinv, bufA);             // agg2 = h2*dinv^2
    k_scatter  <<<blkSC, TB, 0, stream>>>(bufB, ei, norm, bufA);         // agg2 += msgs
    k_bnstats  <<<blkN,  TB, 0, stream>>>(bufA, b2, s2, q2);
    k_bnapply  <<<blkNH, TB, 0, stream>>>(bufA, b2, s2, q2, g2, be2, bufB); // z2

    // --- pool + head ---
    k_pool <<<blkNH, TB, 0, stream>>>(bufB, bat, pooled, counts);
    k_out  <<<1, 64, 0, stream>>>(pooled, counts, Wout, bout, out);

    (void)in_sizes; (void)n_in; (void)out_size; (void)ws_size;
}